// NGram_90812788506978
// MI455X (gfx1250) — compile-verified
//
#include <hip/hip_runtime.h>
#include <hip/hip_bf16.h>
#include <stdint.h>

// N-gram histogram (n=1 and n=2), V=128, L=2048 per row.
// Output row = [128 unigram bins | 16384 bigram bins] as float32.
//
// Data path (all CDNA5-specific):
//   global --async_to_lds--> LDS tokens
//   LDS fp32 bins via ds_add_f32 (exact for small integer counts)
//   LDS --async_from_lds--> global output (no VGPR round trip)

#define NGRAM_V      128
#define NGRAM_LEN    2048
#define NBINS1       128
#define NBINS2       (NGRAM_V * NGRAM_V)     // 16384
#define NB           (NBINS1 + NBINS2)       // 16512
#define TPB          256
#define SMEM_BYTES   ((NB + NGRAM_LEN) * 4)  // 66048 + 8192 = 74240 B

__global__ __launch_bounds__(TPB) void ngram_hist_kernel(
    const int* __restrict__ in, float* __restrict__ out)
{
    extern __shared__ float smem[];
    float* lhist = smem;                 // [NB]  fp32 histogram bins
    int*   ltok  = (int*)(smem + NB);    // [LEN] staged input row

    const int tid = threadIdx.x;
    const int row = blockIdx.x;

    // ---- Stage 1: async global->LDS copy of this row ---------------------
    // 2 x b128 per thread: 256 threads * 2 * 16B = 8192B = full row.
    {
        const uint32_t tb  = (uint32_t)(uintptr_t)ltok;
        const uint64_t gin = (uint64_t)(uintptr_t)(in + (size_t)row * NGRAM_LEN);
        asm volatile("global_load_async_to_lds_b128 %0, %1, off"
                     :: "v"(tb + (uint32_t)tid * 16u),
                        "v"(gin + (uint64_t)tid * 16u) : "memory");
        asm volatile("global_load_async_to_lds_b128 %0, %1, off"
                     :: "v"(tb + (uint32_t)(tid + TPB) * 16u),
                        "v"(gin + (uint64_t)(tid + TPB) * 16u) : "memory");
    }

    // ---- Stage 2: zero the bins while the copy is in flight --------------
    // 0x00000000 is fp32 +0.0, so b128 integer-zero stores are fine.
    for (int i = tid * 4; i < NB; i += TPB * 4) {
        *(int4*)&((int*)lhist)[i] = make_int4(0, 0, 0, 0);
    }

    asm volatile("s_wait_asynccnt 0" ::: "memory");
    __syncthreads();

    // ---- Stage 3: count via LDS fp32 atomics (ds_add_f32) ----------------
    // int4 = tokens [4t..4t+3]: 4 unigrams + 2 aligned bigrams
    // (pairs (0,1),(2,3),...; idx = t0*V + t1, first position most
    //  significant base-V digit, matching the reference).
    {
        const uint32_t hb  = (uint32_t)(uintptr_t)lhist;
        const float    one = 1.0f;
#pragma unroll
        for (int k = 0; k < 2; ++k) {
            const int base = (tid + k * TPB) * 4;
            int4 t = *(const int4*)&ltok[base];
            uint32_t a0 = hb + ((uint32_t)t.x << 2);
            uint32_t a1 = hb + ((uint32_t)t.y << 2);
            uint32_t a2 = hb + ((uint32_t)t.z << 2);
            uint32_t a3 = hb + ((uint32_t)t.w << 2);
            uint32_t b0 = hb + ((uint32_t)(NBINS1 + t.x * NGRAM_V + t.y) << 2);
            uint32_t b1 = hb + ((uint32_t)(NBINS1 + t.z * NGRAM_V + t.w) << 2);
            asm volatile("ds_add_f32 %0, %1" :: "v"(a0), "v"(one) : "memory");
            asm volatile("ds_add_f32 %0, %1" :: "v"(a1), "v"(one) : "memory");
            asm volatile("ds_add_f32 %0, %1" :: "v"(a2), "v"(one) : "memory");
            asm volatile("ds_add_f32 %0, %1" :: "v"(a3), "v"(one) : "memory");
            asm volatile("ds_add_f32 %0, %1" :: "v"(b0), "v"(one) : "memory");
            asm volatile("ds_add_f32 %0, %1" :: "v"(b1), "v"(one) : "memory");
        }
    }

    // Our ds_add_f32 ops are invisible to the compiler's waitcnt pass:
    // drain DScnt explicitly before signaling the barrier.
    asm volatile("s_wait_dscnt 0" ::: "memory");
    __syncthreads();

    // ---- Stage 4: async LDS->global writeback (store-BW bound) -----------
    // Row stride 16512*4 = 66048 B is 16B-aligned -> b128 chunks OK.
    // S_ENDPGM performs an implicit wait-idle, draining ASYNCcnt before the
    // wave (and its LDS) is released.
    {
        const uint32_t hb   = (uint32_t)(uintptr_t)lhist;
        const uint64_t gout = (uint64_t)(uintptr_t)(out + (size_t)row * NB);
        for (int i = tid * 4; i < NB; i += TPB * 4) {
            asm volatile("global_store_async_from_lds_b128 %0, %1, off"
                         :: "v"(gout + ((uint64_t)(uint32_t)i << 2)),
                            "v"(hb + ((uint32_t)i << 2)) : "memory");
        }
    }
}

extern "C" void kernel_launch(void* const* d_in, const int* in_sizes, int n_in,
                              void* d_out, int out_size, void* d_ws, size_t ws_size,
                              hipStream_t stream) {
    (void)n_in; (void)out_size; (void)d_ws; (void)ws_size;
    const int* in = (const int*)d_in[0];
    float* out = (float*)d_out;
    const int batch = in_sizes[0] / NGRAM_LEN;   // 2048

    ngram_hist_kernel<<<dim3(batch), dim3(TPB), SMEM_BYTES, stream>>>(in, out);
}